// EpisodicMemory_67997922230379
// MI455X (gfx1250) — compile-verified
//
#include <hip/hip_runtime.h>
#include <math.h>

typedef float v2f __attribute__((ext_vector_type(2)));
typedef float v4f __attribute__((ext_vector_type(4)));
typedef float v8f __attribute__((ext_vector_type(8)));

#define EPSN 1e-8f
#define TPB 256

// ---------------------------------------------------------------------------
// Kernel 1: inverse row norms for query (B rows) and embeddings (C rows).
// One wave32 per row; shuffle reduction.
// ---------------------------------------------------------------------------
__global__ void rownorm_kernel(const float* __restrict__ q, int B,
                               const float* __restrict__ e, int C, int H,
                               float* __restrict__ invq, float* __restrict__ inve) {
    int wave = (blockIdx.x * blockDim.x + threadIdx.x) >> 5;
    int lane = threadIdx.x & 31;
    int total = B + C;
    if (wave >= total) return;
    const float* src = (wave < B) ? (q + (size_t)wave * H)
                                  : (e + (size_t)(wave - B) * H);
    float s = 0.f;
    for (int h = lane; h < H; h += 32) { float v = src[h]; s += v * v; }
    for (int off = 16; off > 0; off >>= 1) s += __shfl_xor(s, off, 32);
    if (lane == 0) {
        float n = fmaxf(sqrtf(s), EPSN);
        float r = 1.0f / n;
        if (wave < B) invq[wave] = r; else inve[wave - B] = r;
    }
}

// ---------------------------------------------------------------------------
// Kernel 2: sim[b,c] = (q[b]·e[c]) * invq[b] * inve[c] via V_WMMA_F32_16X16X4_F32
// One wave per 16x16 tile. A = Q tile (16xK), B = E tile transposed (Kx16).
// A layout (16x4 f32): lane L: row M = L%16; VGPR0 holds K = 2*(L/16),
// VGPR1 holds K = 2*(L/16)+1  -> contiguous float2 load per lane.
// B layout (4x16 f32) mirrors A with N = L%16, same K split.
// C/D layout: VGPR r: lanes 0-15 -> M=r, lanes 16-31 -> M=r+8; N = lane%16.
// ---------------------------------------------------------------------------
__global__ void sim_wmma_kernel(const float* __restrict__ Q,
                                const float* __restrict__ E,
                                const float* __restrict__ invq,
                                const float* __restrict__ inve,
                                float* __restrict__ sim,
                                int B, int C, int H) {
    int wavesPerBlock = blockDim.x >> 5;
    int waveId = blockIdx.x * wavesPerBlock + (threadIdx.x >> 5);
    int lane = threadIdx.x & 31;
    int tilesN = C >> 4;                 // C divisible by 16
    int tilesM = (B + 15) >> 4;
    if (waveId >= tilesM * tilesN) return;   // whole-wave exit: EXEC stays full
    int tm = waveId / tilesN;
    int tn = waveId % tilesN;

    int halfSel = lane >> 4;             // 0 for lanes 0-15, 1 for lanes 16-31
    int sub     = lane & 15;
    int arow = tm * 16 + sub;            // M row of Q
    if (arow >= B) arow = B - 1;         // clamp loads; stores guarded below
    int brow = tn * 16 + sub;            // N row of E
    int kOfs = halfSel * 2;              // K offset 0 or 2 within each 4-step

    const float* qrow = Q + (size_t)arow * H + kOfs;
    const float* erow = E + (size_t)brow * H + kOfs;

    v8f acc = {};
    for (int k0 = 0; k0 < H; k0 += 4) {
        v2f a = *(const v2f*)(qrow + k0);
        v2f b = *(const v2f*)(erow + k0);
        acc = __builtin_amdgcn_wmma_f32_16x16x4_f32(
            /*neg_a=*/false, a, /*neg_b=*/false, b,
            /*c_mod=*/(short)0, acc, /*reuse_a=*/false, /*reuse_b=*/false);
    }

    int outCol = tn * 16 + sub;
    int mBase  = tm * 16 + halfSel * 8;
    float ie = inve[outCol];
#pragma unroll
    for (int r = 0; r < 8; ++r) {
        int outRow = mBase + r;
        if (outRow < B)
            sim[(size_t)outRow * C + outCol] = acc[r] * invq[outRow] * ie;
    }
}

// ---------------------------------------------------------------------------
// Kernel 3: top-K per row (K<=8). One 256-thread block per row.
// ---------------------------------------------------------------------------
__global__ void topk_kernel(const float* __restrict__ sim, int B, int C, int K,
                            int* __restrict__ topIdx, float* __restrict__ topScores) {
    __shared__ float cs[TPB * 8];
    __shared__ int   ci[TPB * 8];
    __shared__ float rs[TPB];
    __shared__ int   rp[TPB];
    int b = blockIdx.x;
    int t = threadIdx.x;
    const float* row = sim + (size_t)b * C;

    float ls[8]; int li[8];
#pragma unroll
    for (int i = 0; i < 8; ++i) { ls[i] = -INFINITY; li[i] = -1; }
    for (int c = t; c < C; c += TPB) {
        float v = row[c];
        if (v > ls[7]) {
            int p = 7;
            while (p > 0 && v > ls[p - 1]) { ls[p] = ls[p - 1]; li[p] = li[p - 1]; --p; }
            ls[p] = v; li[p] = c;
        }
    }
#pragma unroll
    for (int i = 0; i < 8; ++i) { cs[t * 8 + i] = ls[i]; ci[t * 8 + i] = li[i]; }
    __syncthreads();

    for (int sel = 0; sel < K; ++sel) {
        float best = -INFINITY; int bp = t * 8;
#pragma unroll
        for (int i = 0; i < 8; ++i) {
            float v = cs[t * 8 + i];
            if (v > best) { best = v; bp = t * 8 + i; }
        }
        rs[t] = best; rp[t] = bp;
        __syncthreads();
        for (int off = TPB / 2; off > 0; off >>= 1) {
            if (t < off && rs[t + off] > rs[t]) { rs[t] = rs[t + off]; rp[t] = rp[t + off]; }
            __syncthreads();
        }
        if (t == 0) {
            int p = rp[0];
            topScores[(size_t)b * K + sel] = rs[0];
            topIdx[(size_t)b * K + sel] = ci[p];
            cs[p] = -INFINITY;
        }
        __syncthreads();
    }
}

// ---------------------------------------------------------------------------
// Kernel 4: gather. One block per (b,j) pair; copies S*H floats with B128
// non-temporal loads/stores (streaming ~268 MB; keep it out of L2's hot set).
// Compressed episodes occupy the first CS*H floats of the tile; rest zeros.
// ---------------------------------------------------------------------------
__global__ void gather_kernel(const float* __restrict__ episodes,
                              const float* __restrict__ comp,
                              const unsigned char* __restrict__ isComp,
                              const int* __restrict__ topIdx,
                              float* __restrict__ out,
                              int S, int CS, int H) {
    int pair = blockIdx.x;                       // b*K + j
    int idx = topIdx[pair];
    bool flag = isComp[idx] != 0;
    size_t tile = (size_t)S * H;
    const v4f* srcFull = (const v4f*)(episodes + (size_t)idx * tile);
    const v4f* srcComp = (const v4f*)(comp + (size_t)idx * (size_t)CS * H);
    v4f* dst = (v4f*)(out + (size_t)pair * tile);
    int n4 = (int)(tile >> 2);
    int comp4 = (CS * H) >> 2;
    if (flag) {
        for (int i = threadIdx.x; i < n4; i += blockDim.x) {
            v4f v = (i < comp4) ? __builtin_nontemporal_load(srcComp + i)
                                : (v4f){0.f, 0.f, 0.f, 0.f};
            __builtin_nontemporal_store(v, dst + i);
        }
    } else {
        for (int i = threadIdx.x; i < n4; i += blockDim.x) {
            v4f v = __builtin_nontemporal_load(srcFull + i);
            __builtin_nontemporal_store(v, dst + i);
        }
    }
}

// ---------------------------------------------------------------------------
extern "C" void kernel_launch(void* const* d_in, const int* in_sizes, int n_in,
                              void* d_out, int out_size, void* d_ws, size_t ws_size,
                              hipStream_t stream) {
    const float* query    = (const float*)d_in[0];
    const float* emb      = (const float*)d_in[1];
    const float* episodes = (const float*)d_in[2];
    const float* comp     = (const float*)d_in[3];
    const unsigned char* isComp = (const unsigned char*)d_in[4];

    // Derive shapes from flat sizes (B=256,H=256,C=16384,S=64,CS=32,K=8 for ref).
    long long C  = in_sizes[4];
    long long H  = in_sizes[1] / C;
    long long B  = in_sizes[0] / H;
    long long S  = in_sizes[2] / (C * H);
    long long CS = in_sizes[3] / (C * H);
    long long K  = (long long)out_size / (B * (S * H + 1));

    float* out = (float*)d_out;
    float* scoresOut = out + (size_t)(B * K * S * H);   // tuple tail

    // Workspace: invq[B] | inve[C] | sim[B*C] | topIdx[B*K]
    float* invq = (float*)d_ws;
    float* inve = invq + B;
    float* sim  = inve + C;
    int* topIdx = (int*)(sim + (size_t)B * C);

    // 1) inverse norms: one wave per row, 8 waves per block
    {
        long long rows = B + C;
        long long blocks = (rows + 7) / 8;
        rownorm_kernel<<<(int)blocks, 256, 0, stream>>>(query, (int)B, emb, (int)C,
                                                        (int)H, invq, inve);
    }
    // 2) similarity GEMM with f32 WMMA: one wave per 16x16 tile, 8 waves/block
    {
        long long tiles = ((B + 15) / 16) * (C / 16);
        long long blocks = (tiles + 7) / 8;
        sim_wmma_kernel<<<(int)blocks, 256, 0, stream>>>(query, emb, invq, inve, sim,
                                                         (int)B, (int)C, (int)H);
    }
    // 3) top-K per row
    topk_kernel<<<(int)B, TPB, 0, stream>>>(sim, (int)B, (int)C, (int)K,
                                            topIdx, scoresOut);
    // 4) bandwidth-bound gather (dominant cost: ~268 MB -> ~11.5 us at 23.3 TB/s)
    gather_kernel<<<(int)(B * K), TPB, 0, stream>>>(episodes, comp, isComp, topIdx,
                                                    out, (int)S, (int)CS, (int)H);
}